// SparseLinear_72679436582939
// MI455X (gfx1250) — compile-verified
//
#include <hip/hip_runtime.h>
#include <stdint.h>

// ---------------------------------------------------------------------------
// SparseLinear: out[b,m] = sum_{e: dst[e]==m} x[b,src[e]] * val[e] + bias[m]
// B = 32 (== wave32 lanes), N = M = 100000, E = 3.2M.
// Memory-bound scatter/gather: working set ~64MB fits in MI455X 192MB L2.
// Strategy:
//   1) transpose x (B,N) -> x_t (N,32): per-edge gather = 1 contiguous 128B line
//   2) zero out_t (M,32)
//   3) edge kernel: lane = batch; edge stream double-buffered into LDS with
//      CDNA5 global_load_async_to_lds_b128 + s_wait_asynccnt; coalesced
//      wave-wide global_atomic_add_f32 (one 128B line per edge)
//   4) finalize: out(32,M) = out_t(M,32)^T + bias via v_wmma_f32_16x16x4_f32
//      (D = I * S + C_bias is bit-exact; converts layout so stores coalesce)
// ---------------------------------------------------------------------------

#define WPB    8      // waves per 256-thread block
#define EPW    128    // edges per wave
#define CHUNK  32     // edges per async chunk (src+dst+val = 3*128B)
#define BUFB   512    // LDS bytes per chunk buffer (src|dst|val|scratch)

#define __AS1 __attribute__((address_space(1)))
#define __AS3 __attribute__((address_space(3)))

typedef __attribute__((ext_vector_type(2))) float v2f;
typedef __attribute__((ext_vector_type(8))) float v8f;
typedef __attribute__((ext_vector_type(4))) int   v4i;

#if defined(__gfx1250__) && __has_builtin(__builtin_amdgcn_global_load_async_to_lds_b128)
#define HAVE_ASYNC_LDS 1
#else
#define HAVE_ASYNC_LDS 0
#endif

#if defined(__gfx1250__) && __has_builtin(__builtin_amdgcn_wmma_f32_16x16x4_f32)
#define HAVE_WMMA_F32 1
#else
#define HAVE_WMMA_F32 0
#endif

__device__ __forceinline__ void wait_async_le(int n) {
#if defined(__gfx1250__)
#if __has_builtin(__builtin_amdgcn_s_wait_asynccnt)
  if (n) __builtin_amdgcn_s_wait_asynccnt(1);
  else   __builtin_amdgcn_s_wait_asynccnt(0);
#else
  if (n) asm volatile("s_wait_asynccnt 0x1" ::: "memory");
  else   asm volatile("s_wait_asynccnt 0x0" ::: "memory");
#endif
#endif
}

// ------------------------------- zero out_t --------------------------------
__global__ void sl_zero(float4* __restrict__ p, long n4) {
  long i = (long)blockIdx.x * blockDim.x + threadIdx.x;
  long stride = (long)gridDim.x * blockDim.x;
  for (; i < n4; i += stride) p[i] = make_float4(0.f, 0.f, 0.f, 0.f);
}

// --------------------------- transpose x -> x_t ----------------------------
// x: (32, N) row-major -> x_t: (N, 32). 32x32 LDS tiles, both sides coalesced.
__global__ __launch_bounds__(256) void sl_transpose_x(
    const float* __restrict__ x, float* __restrict__ xt, int N) {
  __shared__ float tile[32][33];
  const int n0 = blockIdx.x * 32;
  const int tx = threadIdx.x & 31;
  const int ty = threadIdx.x >> 5;  // 0..7
#pragma unroll
  for (int i = 0; i < 4; ++i) {
    int b = ty + i * 8;
    int n = n0 + tx;
    float v = (n < N) ? x[(long)b * N + n] : 0.f;
    tile[tx][b] = v;
  }
  __syncthreads();
#pragma unroll
  for (int i = 0; i < 4; ++i) {
    int r = ty + i * 8;
    int n = n0 + r;
    if (n < N) xt[(long)n * 32 + tx] = tile[r][tx];
  }
}

// ------------------------------ edge scatter -------------------------------
#if HAVE_ASYNC_LDS
// One instruction fetches a whole 32-edge chunk: lanes 0-7 src, 8-15 dst,
// 16-23 val (16B each); lanes 24-31 land in a scratch slot (kept busy with
// valid addresses so EXEC stays all-ones).
__device__ __forceinline__ void issue_chunk_async(
    const int* __restrict__ src, const int* __restrict__ dst,
    const float* __restrict__ val, long e0, char* lbuf, int lane) {
  const int g = lane >> 3;   // 0:src 1:dst 2:val 3:dup-val->scratch
  const int j = lane & 7;
  const char* gp;
  if (g == 0)      gp = (const char*)(src + e0) + j * 16;
  else if (g == 1) gp = (const char*)(dst + e0) + j * 16;
  else             gp = (const char*)(val + e0) + j * 16;
  char* lp = lbuf + g * 128 + j * 16;
  __builtin_amdgcn_global_load_async_to_lds_b128(
      (__AS1 v4i*)gp, (__AS3 v4i*)lp, 0, 0);
}

__device__ __forceinline__ void process_chunk_lds(
    const char* lbuf, int lane, const float* __restrict__ xt,
    float* __restrict__ outt) {
  const int*   ls = (const int*)(lbuf);
  const int*   ld = (const int*)(lbuf + 128);
  const float* lv = (const float*)(lbuf + 256);
#pragma unroll 4
  for (int k = 0; k < CHUNK; ++k) {
    int   s = ls[k];   // uniform-address ds_load -> broadcast
    int   d = ld[k];
    float v = lv[k];
    float xv = xt[(long)s * 32 + lane];                 // one 128B line
    atomicAdd(outt + (long)d * 32 + lane, xv * v);      // one 128B line
  }
}
#endif

__global__ __launch_bounds__(256) void sl_edges(
    const int* __restrict__ src, const int* __restrict__ dst,
    const float* __restrict__ val, const float* __restrict__ xt,
    float* __restrict__ outt, long E) {
  const int wid  = threadIdx.x >> 5;
  const int lane = threadIdx.x & 31;
  const long wg = (long)blockIdx.x * WPB + wid;
  const long e_begin = wg * EPW;
  if (e_begin >= E) return;
  long e_end = e_begin + EPW;
  if (e_end > E) e_end = E;

#if HAVE_ASYNC_LDS
  __shared__ char lds[WPB * 2 * BUFB];
  char* buf = lds + wid * (2 * BUFB);
  const int nch = (int)((e_end - e_begin + CHUNK - 1) / CHUNK);
  if (e_begin + CHUNK <= E)
    issue_chunk_async(src, dst, val, e_begin, buf, lane);
  for (int c = 0; c < nch; ++c) {
    const long e0 = e_begin + (long)c * CHUNK;
    const bool cur_full = (e0 + CHUNK <= E);
    const long e1 = e0 + CHUNK;
    const bool nxt_full = (e1 < e_end) && (e1 + CHUNK <= E);
    if (nxt_full)
      issue_chunk_async(src, dst, val, e1, buf + ((c + 1) & 1) * BUFB, lane);
    if (cur_full) {
      wait_async_le(nxt_full ? 1 : 0);
      process_chunk_lds(buf + (c & 1) * BUFB, lane, xt, outt);
    } else {
      const int cnt = (int)(e_end - e0);
      for (int k = 0; k < cnt; ++k) {
        int s = src[e0 + k]; int d = dst[e0 + k]; float v = val[e0 + k];
        atomicAdd(outt + (long)d * 32 + lane, xt[(long)s * 32 + lane] * v);
      }
    }
  }
#else
  // Fallback: per-lane edge loads + wave broadcast via shuffles.
  for (long e0 = e_begin; e0 < e_end; e0 += 32) {
    long ei = e0 + lane;
    int s = 0, d = 0; float v = 0.f;
    if (ei < E) { s = src[ei]; d = dst[ei]; v = val[ei]; }
    int cnt = (int)((e_end - e0 < 32) ? (e_end - e0) : 32);
    for (int k = 0; k < cnt; ++k) {
      int   sk = __shfl(s, k, 32);
      int   dk = __shfl(d, k, 32);
      float vk = __shfl(v, k, 32);
      atomicAdd(outt + (long)dk * 32 + lane, xt[(long)sk * 32 + lane] * vk);
    }
  }
#endif
}

// ------------------------------- finalize ----------------------------------
// out[b][m] = out_t[m][b] + bias[m].  One wave per 16-row m-tile; both
// versions use identical launch geometry (host can't see device __has_builtin).
#if HAVE_WMMA_F32
__global__ __launch_bounds__(256) void sl_finalize(
    const float* __restrict__ outt, const float* __restrict__ bias,
    float* __restrict__ out, int M) {
  const int wid  = threadIdx.x >> 5;
  const int lane = threadIdx.x & 31;
  const int m0 = (blockIdx.x * WPB + wid) * 16;
  if (m0 >= M) return;                  // wave-uniform; EXEC stays all-ones
  const int col = lane & 15;            // N index (m offset) in B/C/D layouts
  const int hi  = lane >> 4;            // 0/1
  const float bv = bias[m0 + col];      // C[M][N] = bias[m0+N] for all rows
#pragma unroll
  for (int bt = 0; bt < 2; ++bt) {      // b in [0,16) and [16,32)
    v8f acc;
#pragma unroll
    for (int r = 0; r < 8; ++r) acc[r] = bv;
#pragma unroll
    for (int c = 0; c < 4; ++c) {       // K-chunks: D = sum_c I_c * S_c + C
      v2f a, b;
      // A = identity slice: A[i][k] = (i == c*4+k); lane row = col, K = 2*hi(+1)
      a.x = (col == c * 4 + 2 * hi)     ? 1.0f : 0.0f;
      a.y = (col == c * 4 + 2 * hi + 1) ? 1.0f : 0.0f;
      // B = S_c where S[j][i] = out_t[(m0+i)*32 + bt*16 + j]
      const float* bp = outt + (long)(m0 + col) * 32 + bt * 16 + c * 4 + 2 * hi;
      b.x = bp[0];
      b.y = bp[1];
      acc = __builtin_amdgcn_wmma_f32_16x16x4_f32(
          false, a, false, b, (short)0, acc, false, false);
    }
    // D layout: vgpr r, lane l -> row M=r+8*hi, col N=col; row here is b.
#pragma unroll
    for (int r = 0; r < 8; ++r) {
      int brow = bt * 16 + r + 8 * hi;
      out[(long)brow * M + m0 + col] = acc[r];   // 2x 64B segments per store
    }
  }
}
#else
__global__ __launch_bounds__(256) void sl_finalize(
    const float* __restrict__ outt, const float* __restrict__ bias,
    float* __restrict__ out, int M) {
  __shared__ float tile[WPB][16][33];
  const int wid  = threadIdx.x >> 5;
  const int lane = threadIdx.x & 31;
  const int m0 = (blockIdx.x * WPB + wid) * 16;
  if (m0 >= M) return;
  float(*t)[33] = tile[wid];
#pragma unroll
  for (int i = 0; i < 16; ++i) t[i][lane] = outt[(long)(m0 + i) * 32 + lane];
  // wave-private tile: per-wave LDS ops are in-order, no barrier needed
  const int col = lane & 15;
  const int hi  = lane >> 4;
  const float bv = bias[m0 + col];
  for (int bb = hi; bb < 32; bb += 2)
    out[(long)bb * M + m0 + col] = t[col][bb] + bv;
}
#endif

// ------------------------- generic fallback path ---------------------------
__global__ void sl_bias_init(float* __restrict__ out,
                             const float* __restrict__ bias, long B, long M) {
  long i = (long)blockIdx.x * blockDim.x + threadIdx.x;
  long stride = (long)gridDim.x * blockDim.x;
  long total = B * M;
  for (; i < total; i += stride) out[i] = bias[i % M];
}

__global__ void sl_edges_direct(
    const int* __restrict__ src, const int* __restrict__ dst,
    const float* __restrict__ val, const float* __restrict__ x,
    float* __restrict__ out, long E, int N, int M, int B) {
  long e = (long)blockIdx.x * blockDim.x + threadIdx.x;
  if (e >= E) return;
  int s = src[e], d = dst[e];
  float v = val[e];
  for (int b = 0; b < B; ++b)
    atomicAdd(out + (long)b * M + d, x[(long)b * N + s] * v);
}

// ------------------------------ host launcher ------------------------------
extern "C" void kernel_launch(void* const* d_in, const int* in_sizes, int n_in,
                              void* d_out, int out_size, void* d_ws,
                              size_t ws_size, hipStream_t stream) {
  const float* x    = (const float*)d_in[0];
  const int*   idx  = (const int*)d_in[1];
  const float* val  = (const float*)d_in[2];
  const float* bias = (const float*)d_in[3];
  const long E = in_sizes[2];
  const long M = in_sizes[3];
  const long B = (long)out_size / M;
  const long N = in_sizes[0] / B;
  const int* src = idx;
  const int* dst = idx + E;
  float* out = (float*)d_out;

  const size_t need = (size_t)(N + M) * 32 * sizeof(float);
  if (B == 32 && (M % 16 == 0) && ws_size >= need) {
    float* xt   = (float*)d_ws;
    float* outt = xt + N * 32;
    sl_zero<<<1024, 256, 0, stream>>>((float4*)outt, M * 8);
    sl_transpose_x<<<dim3((unsigned)((N + 31) / 32)), 256, 0, stream>>>(
        x, xt, (int)N);
    const long waves  = (E + EPW - 1) / EPW;
    const long blocks = (waves + WPB - 1) / WPB;
    sl_edges<<<dim3((unsigned)blocks), 256, 0, stream>>>(src, dst, val, xt,
                                                         outt, E);
    const long tiles   = M / 16;
    const long fblocks = (tiles + WPB - 1) / WPB;
    sl_finalize<<<dim3((unsigned)fblocks), 256, 0, stream>>>(outt, bias, out,
                                                             (int)M);
  } else {
    sl_bias_init<<<1024, 256, 0, stream>>>(out, bias, B, M);
    sl_edges_direct<<<dim3((unsigned)((E + 255) / 256)), 256, 0, stream>>>(
        src, dst, val, x, out, E, (int)N, (int)M, (int)B);
  }
}